// Net_coor_res_85478439125047
// MI455X (gfx1250) — compile-verified
//
#include <hip/hip_runtime.h>
#include <math.h>

// Problem constants (match reference)
#define NNODES 30000
#define NEDGES 300000
#define FDIM   64
#define DDIM   128
#define HEADS  8
#define CH     16
#define EDIM   16
#define NLAYER 3

typedef __attribute__((ext_vector_type(2))) float v2f;
typedef __attribute__((ext_vector_type(8))) float v8f;

__device__ __forceinline__ float elu_f(float x) {
    return x > 0.f ? x : (expf(x) - 1.f);
}
__device__ __forceinline__ float gelu_f(float x) {
    return 0.5f * x * (1.f + erff(x * 0.70710678118654752f));
}

// Monotonic uint encoding of float for atomicMax-based segment max
__device__ __forceinline__ unsigned ford(float f) {
    unsigned u = __float_as_uint(f);
    return (u & 0x80000000u) ? ~u : (u | 0x80000000u);
}
__device__ __forceinline__ float fdec(unsigned u) {
    return (u & 0x80000000u) ? __uint_as_float(u & 0x7fffffffu)
                             : __uint_as_float(~u);
}

// ---------------------------------------------------------------------------
// Fused GEMM: out[N,128] = act(A[N,K] @ W^T + bias) (+ res)
// W row-major [128, K]. One wave computes one 16x16 tile via
// V_WMMA_F32_16X16X4_F32 (fp32, K-step 4). 8 waves/block = full Dout row.
// act: 0=none, 1=elu, 2=gelu
// ---------------------------------------------------------------------------
__global__ void __launch_bounds__(256)
gemm128_wmma(const float* __restrict__ A, const float* __restrict__ W,
             const float* __restrict__ bias, const float* __restrict__ res,
             float* __restrict__ out, int K, int act)
{
    const int lane = threadIdx.x & 31;
    const int wave = threadIdx.x >> 5;
    const int m0   = blockIdx.x * 16;
    const int n0   = wave * 16;
    const int half = lane >> 4;   // 0: K+0/K+1, 1: K+2/K+3
    const int l15  = lane & 15;

    v8f acc = {0.f, 0.f, 0.f, 0.f, 0.f, 0.f, 0.f, 0.f};

    const float* ap = A + (size_t)(m0 + l15) * K + half * 2;
    const float* bp = W + (size_t)(n0 + l15) * K + half * 2;

    for (int k = 0; k < K; k += 4) {
        v2f a = *reinterpret_cast<const v2f*>(ap + k);
        v2f b = *reinterpret_cast<const v2f*>(bp + k);
        acc = __builtin_amdgcn_wmma_f32_16x16x4_f32(
            /*neg_a=*/false, a, /*neg_b=*/false, b,
            /*c_mod=*/(short)0, acc, /*reuse_a=*/false, /*reuse_b=*/false);
    }

#pragma unroll
    for (int g = 0; g < 8; ++g) {
        const int row = m0 + g + half * 8;
        const int col = n0 + l15;
        float v = acc[g] + bias[col];
        if (act == 1)      v = elu_f(v);
        else if (act == 2) v = gelu_f(v);
        if (res) v += res[(size_t)row * DDIM + col];
        out[(size_t)row * DDIM + col] = v;
    }
}

// ---------------------------------------------------------------------------
// LayerNorm over D=128: one wave per row, 4 elems/lane, shfl_xor reduce.
// ---------------------------------------------------------------------------
__global__ void __launch_bounds__(256)
ln_kernel(const float* __restrict__ x, const float* __restrict__ g,
          const float* __restrict__ b, float* __restrict__ out, int nrows)
{
    const int lane = threadIdx.x & 31;
    const int wave = threadIdx.x >> 5;
    const int row  = blockIdx.x * 8 + wave;
    if (row >= nrows) return;

    const float4 v = *reinterpret_cast<const float4*>(x + (size_t)row * DDIM + lane * 4);
    float s  = v.x + v.y + v.z + v.w;
    float s2 = v.x * v.x + v.y * v.y + v.z * v.z + v.w * v.w;
#pragma unroll
    for (int off = 16; off >= 1; off >>= 1) {
        s  += __shfl_xor(s,  off, 32);
        s2 += __shfl_xor(s2, off, 32);
    }
    const float mu  = s * (1.f / 128.f);
    const float var = s2 * (1.f / 128.f) - mu * mu;
    const float inv = rsqrtf(var + 1e-5f);

    const float4 gg = *reinterpret_cast<const float4*>(g + lane * 4);
    const float4 bb = *reinterpret_cast<const float4*>(b + lane * 4);
    float4 o;
    o.x = (v.x - mu) * inv * gg.x + bb.x;
    o.y = (v.y - mu) * inv * gg.y + bb.y;
    o.z = (v.z - mu) * inv * gg.z + bb.z;
    o.w = (v.w - mu) * inv * gg.w + bb.w;
    *reinterpret_cast<float4*>(out + (size_t)row * DDIM + lane * 4) = o;
}

// ---------------------------------------------------------------------------
// Init segment-softmax state: m (ordered-uint, 0 == -inf key), denom = 0
// ---------------------------------------------------------------------------
__global__ void init_softmax_kernel(unsigned* __restrict__ mbuf,
                                    float* __restrict__ den)
{
    const int tid = blockIdx.x * blockDim.x + threadIdx.x;
    if (tid < NNODES * HEADS) {
        mbuf[tid] = 0u;
        den[tid]  = 0.f;
    }
}

// ---------------------------------------------------------------------------
// Pass E1: alpha[e,h] = (q[dst]·(k[src]+e))/4 ; atomicMax into m[dst,h]
// One wave per edge, 4 channels/lane. e computed on the fly (ew in LDS).
// ---------------------------------------------------------------------------
__global__ void __launch_bounds__(256)
edge_alpha_kernel(const int* __restrict__ ei, const float* __restrict__ ea,
                  const float* __restrict__ q, const float* __restrict__ k,
                  const float* __restrict__ ew, float* __restrict__ alpha,
                  unsigned* __restrict__ mbuf)
{
    __shared__ float sw[DDIM * EDIM];
    for (int t = threadIdx.x; t < DDIM * EDIM; t += blockDim.x) sw[t] = ew[t];
    __syncthreads();

    const int lane = threadIdx.x & 31;
    const int wave = threadIdx.x >> 5;
    const int e    = blockIdx.x * 8 + wave;
    if (e >= NEDGES) return;

    const int src = ei[e];
    const int dst = ei[NEDGES + e];

    const float eav = (lane < EDIM) ? ea[(size_t)e * EDIM + lane] : 0.f;
    const int   c0  = lane * 4;

    float ec[4] = {0.f, 0.f, 0.f, 0.f};
    for (int t = 0; t < EDIM; ++t) {
        const float a = __shfl(eav, t, 32);
#pragma unroll
        for (int j = 0; j < 4; ++j) ec[j] += a * sw[(c0 + j) * EDIM + t];
    }

    const float4 qv = *reinterpret_cast<const float4*>(q + (size_t)dst * DDIM + c0);
    const float4 kv = *reinterpret_cast<const float4*>(k + (size_t)src * DDIM + c0);
    float p = qv.x * (kv.x + ec[0]) + qv.y * (kv.y + ec[1]) +
              qv.z * (kv.z + ec[2]) + qv.w * (kv.w + ec[3]);
    p += __shfl_xor(p, 1, 32);
    p += __shfl_xor(p, 2, 32);   // reduce 4-lane head group (C=16)

    if ((lane & 3) == 0) {
        const int h = lane >> 2;
        const float a = p * 0.25f;   // 1/sqrt(C), C=16
        alpha[(size_t)e * HEADS + h] = a;
        atomicMax(&mbuf[(size_t)dst * HEADS + h], ford(a));
    }
}

// ---------------------------------------------------------------------------
// Pass E2: alpha = exp(alpha - m[dst]); denom[dst] += alpha
// ---------------------------------------------------------------------------
__global__ void edge_softmax_kernel(const int* __restrict__ dsti,
                                    float* __restrict__ alpha,
                                    const unsigned* __restrict__ mbuf,
                                    float* __restrict__ den)
{
    const int tid = blockIdx.x * blockDim.x + threadIdx.x;
    if (tid >= NEDGES * HEADS) return;
    const int e = tid >> 3, h = tid & 7;
    const int dst = dsti[e];
    const float m  = fdec(mbuf[(size_t)dst * HEADS + h]);
    const float ex = expf(alpha[tid] - m);
    alpha[tid] = ex;
    __hip_atomic_fetch_add(&den[(size_t)dst * HEADS + h], ex,
                           __ATOMIC_RELAXED, __HIP_MEMORY_SCOPE_AGENT);
}

// ---------------------------------------------------------------------------
// Pass E3: att[dst] += (alpha/denom) * (v[src] + e). One wave per edge.
// ---------------------------------------------------------------------------
__global__ void __launch_bounds__(256)
edge_message_kernel(const int* __restrict__ ei, const float* __restrict__ ea,
                    const float* __restrict__ v, const float* __restrict__ ew,
                    const float* __restrict__ alpha, const float* __restrict__ den,
                    float* __restrict__ att)
{
    __shared__ float sw[DDIM * EDIM];
    for (int t = threadIdx.x; t < DDIM * EDIM; t += blockDim.x) sw[t] = ew[t];
    __syncthreads();

    const int lane = threadIdx.x & 31;
    const int wave = threadIdx.x >> 5;
    const int e    = blockIdx.x * 8 + wave;
    if (e >= NEDGES) return;

    const int src = ei[e];
    const int dst = ei[NEDGES + e];
    const int h   = lane >> 2;

    const float ex   = alpha[(size_t)e * HEADS + h];
    const float dn   = den[(size_t)dst * HEADS + h];
    const float coef = ex / (dn + 1e-16f);

    const float eav = (lane < EDIM) ? ea[(size_t)e * EDIM + lane] : 0.f;
    const int   c0  = lane * 4;

    float ec[4] = {0.f, 0.f, 0.f, 0.f};
    for (int t = 0; t < EDIM; ++t) {
        const float a = __shfl(eav, t, 32);
#pragma unroll
        for (int j = 0; j < 4; ++j) ec[j] += a * sw[(c0 + j) * EDIM + t];
    }

    const float4 vv = *reinterpret_cast<const float4*>(v + (size_t)src * DDIM + c0);
    float* o = att + (size_t)dst * DDIM + c0;
    __hip_atomic_fetch_add(&o[0], coef * (vv.x + ec[0]), __ATOMIC_RELAXED, __HIP_MEMORY_SCOPE_AGENT);
    __hip_atomic_fetch_add(&o[1], coef * (vv.y + ec[1]), __ATOMIC_RELAXED, __HIP_MEMORY_SCOPE_AGENT);
    __hip_atomic_fetch_add(&o[2], coef * (vv.z + ec[2]), __ATOMIC_RELAXED, __HIP_MEMORY_SCOPE_AGENT);
    __hip_atomic_fetch_add(&o[3], coef * (vv.w + ec[3]), __ATOMIC_RELAXED, __HIP_MEMORY_SCOPE_AGENT);
}

// ---------------------------------------------------------------------------
// Output head: out[N,3] = hn @ linl_w.T + linl_b
// ---------------------------------------------------------------------------
__global__ void head_kernel(const float* __restrict__ hn,
                            const float* __restrict__ w,
                            const float* __restrict__ b,
                            float* __restrict__ out)
{
    const int n = blockIdx.x * blockDim.x + threadIdx.x;
    if (n >= NNODES) return;
    const float* xr = hn + (size_t)n * DDIM;
#pragma unroll
    for (int j = 0; j < 3; ++j) {
        float s = b[j];
        const float* wr = w + j * DDIM;
        for (int d = 0; d < DDIM; ++d) s += xr[d] * wr[d];
        out[(size_t)n * 3 + j] = s;
    }
}

// ---------------------------------------------------------------------------
extern "C" void kernel_launch(void* const* d_in, const int* in_sizes, int n_in,
                              void* d_out, int out_size, void* d_ws, size_t ws_size,
                              hipStream_t stream)
{
    const float* x       = (const float*)d_in[0];
    const int*   ei      = (const int*)  d_in[1];   // [2,E]: row0=src, row1=dst
    const float* ea      = (const float*)d_in[2];
    const float* lin0_w  = (const float*)d_in[3];
    const float* lin0_b  = (const float*)d_in[4];
    const float* q_w     = (const float*)d_in[5];
    const float* q_b     = (const float*)d_in[6];
    const float* k_w     = (const float*)d_in[7];
    const float* k_b     = (const float*)d_in[8];
    const float* v_w     = (const float*)d_in[9];
    const float* v_b     = (const float*)d_in[10];
    const float* e_w     = (const float*)d_in[11];
    const float* skip_w  = (const float*)d_in[12];
    const float* skip_b  = (const float*)d_in[13];
    const float* ln1_g   = (const float*)d_in[14];
    const float* ln1_b   = (const float*)d_in[15];
    const float* lins_w  = (const float*)d_in[16];
    const float* lins_b  = (const float*)d_in[17];
    const float* ln2_g   = (const float*)d_in[18];
    const float* ln2_b   = (const float*)d_in[19];
    const float* lins2_w = (const float*)d_in[20];
    const float* lins2_b = (const float*)d_in[21];
    const float* lnl_g   = (const float*)d_in[22];
    const float* lnl_b   = (const float*)d_in[23];
    const float* linl_w  = (const float*)d_in[24];
    const float* linl_b  = (const float*)d_in[25];
    float* out = (float*)d_out;

    // Workspace layout
    float* ws = (float*)d_ws;
    const size_t nd = (size_t)NNODES * DDIM;
    float*    H0    = ws;            // current hidden state
    float*    B1    = ws + 1 * nd;   // LN output / identity
    float*    Q     = ws + 2 * nd;
    float*    Kb    = ws + 3 * nd;
    float*    V     = ws + 4 * nd;
    float*    ATT   = ws + 5 * nd;   // skip + aggregated messages
    float*    ALPHA = ws + 6 * nd;                         // [E,H]
    unsigned* MB    = (unsigned*)(ALPHA + (size_t)NEDGES * HEADS); // [N,H]
    float*    DEN   = (float*)(MB + (size_t)NNODES * HEADS);       // [N,H]

    const int gGemm = NNODES / 16;                 // 1875
    const int gLN   = (NNODES + 7) / 8;
    const int gEdge = (NEDGES + 7) / 8;
    const int gEH   = (NEDGES * HEADS + 255) / 256;
    const int gNH   = (NNODES * HEADS + 255) / 256;
    const int gNode = (NNODES + 255) / 256;

    // Input projection: h = elu(x @ lin0_w.T + b)
    gemm128_wmma<<<gGemm, 256, 0, stream>>>(x, lin0_w, lin0_b, nullptr, H0, FDIM, 1);

    for (int i = 0; i < NLAYER; ++i) {
        const size_t dd = (size_t)i * DDIM * DDIM;
        const size_t d1 = (size_t)i * DDIM;
        const float* ewl = e_w + (size_t)i * DDIM * EDIM;

        // hn = LN(h)
        ln_kernel<<<gLN, 256, 0, stream>>>(H0, ln1_g + d1, ln1_b + d1, B1, NNODES);

        // q/k/v/skip projections of hn
        gemm128_wmma<<<gGemm, 256, 0, stream>>>(B1, q_w + dd,    q_b + d1,    nullptr, Q,   DDIM, 0);
        gemm128_wmma<<<gGemm, 256, 0, stream>>>(B1, k_w + dd,    k_b + d1,    nullptr, Kb,  DDIM, 0);
        gemm128_wmma<<<gGemm, 256, 0, stream>>>(B1, v_w + dd,    v_b + d1,    nullptr, V,   DDIM, 0);
        gemm128_wmma<<<gGemm, 256, 0, stream>>>(B1, skip_w + dd, skip_b + d1, nullptr, ATT, DDIM, 0);

        // segment softmax + message aggregation into ATT
        init_softmax_kernel<<<gNH, 256, 0, stream>>>(MB, DEN);
        edge_alpha_kernel<<<gEdge, 256, 0, stream>>>(ei, ea, Q, Kb, ewl, ALPHA, MB);
        edge_softmax_kernel<<<gEH, 256, 0, stream>>>(ei + NEDGES, ALPHA, MB, DEN);
        edge_message_kernel<<<gEdge, 256, 0, stream>>>(ei, ea, V, ewl, ALPHA, DEN, ATT);

        // identity = gelu(ATT @ lins_w.T + b) + h_in   -> B1
        gemm128_wmma<<<gGemm, 256, 0, stream>>>(ATT, lins_w + dd, lins_b + d1, H0, B1, DDIM, 2);
        // tmp = LN(identity) -> Q (free)
        ln_kernel<<<gLN, 256, 0, stream>>>(B1, ln2_g + d1, ln2_b + d1, Q, NNODES);
        // h = elu(tmp @ lins2_w.T + b) + identity      -> H0
        gemm128_wmma<<<gGemm, 256, 0, stream>>>(Q, lins2_w + dd, lins2_b + d1, B1, H0, DDIM, 1);
    }

    // Final LN + 3-way linear head
    ln_kernel<<<gLN, 256, 0, stream>>>(H0, lnl_g, lnl_b, B1, NNODES);
    head_kernel<<<gNode, 256, 0, stream>>>(B1, linl_w, linl_b, out);
}